// ModeSelection_82952998355466
// MI455X (gfx1250) — compile-verified
//
#include <hip/hip_runtime.h>

typedef __attribute__((ext_vector_type(2))) float v2f;
typedef __attribute__((ext_vector_type(8))) float v8f;

#define NB 4096
#define NP 10
#define NLD 512
#define NTD 512
#define NDIN 2048
#define MROWS (NB * NP)          // 40960
#define K2 (NLD * NP)            // 5120
#define LDS_STRIDE 68            // 64 + 4 pad -> conflict-free ds_load_b64

// ---------------------------------------------------------------------------
// Kernel 1: emb[m, n] = relu( ctx[m, :] @ W1[:, n] + b1[n] )
// ctx row m = [agent[b] | lane[b,p] | nghl[b,p] | ngh[b,p]],  b=m/10, p=m%10
// Block: 8 waves = 64 M-rows x 128 N-cols. Wave: 4 subtiles (64Mx16N).
// ---------------------------------------------------------------------------
__global__ __launch_bounds__(256) void k1_gemm1_relu(
    const float* __restrict__ agent,   // [B,512]
    const float* __restrict__ lane,    // [B,10,512]
    const float* __restrict__ nghl,    // [B,10,512]
    const float* __restrict__ ngh,     // [B,10,512]
    const float* __restrict__ W1,      // [2048,512] row-major
    const float* __restrict__ b1,      // [512]
    float* __restrict__ emb)           // [40960,512]
{
    __shared__ float ldsA[64 * LDS_STRIDE];

    const int tid     = threadIdx.x;
    const int wave    = tid >> 5;
    const int lane_id = tid & 31;
    const int khalf   = lane_id >> 4;        // 0: K pair (0,1), 1: K pair (2,3)
    const int l15     = lane_id & 15;
    const int M0      = blockIdx.x * 64;
    const int ncol    = blockIdx.y * 128 + wave * 16 + l15;

    v8f c0 = {}, c1 = {}, c2 = {}, c3 = {};

    for (int chunk = 0; chunk < 32; ++chunk) {
        const int kc     = chunk * 64;       // global K base of this chunk
        const int region = chunk >> 3;       // which concat source (0..3)
        const int koff   = kc & 511;         // offset inside that source

        // Cooperative stage of A chunk: 64 rows x 64 K = 1024 float4 loads.
        #pragma unroll
        for (int j = 0; j < 4; ++j) {
            const int i   = tid + j * 256;   // 0..1023
            const int row = i >> 4;
            const int kq  = (i & 15) << 2;
            const int m   = M0 + row;
            const int b   = m / 10;
            const int p   = m - b * 10;
            const float* s;
            if (region == 0)      s = agent + (size_t)b * 512 + koff + kq;
            else if (region == 1) s = lane  + ((size_t)b * 10 + p) * 512 + koff + kq;
            else if (region == 2) s = nghl  + ((size_t)b * 10 + p) * 512 + koff + kq;
            else                  s = ngh   + ((size_t)b * 10 + p) * 512 + koff + kq;
            *(float4*)&ldsA[row * LDS_STRIDE + kq] = *(const float4*)s;
        }
        __syncthreads();

        #pragma unroll 4
        for (int kk = 0; kk < 64; kk += 4) {
            // B fragment: W1[kb..kb+1, ncol]  (stride-512 dword loads, L2-hot)
            const int kb = kc + kk + 2 * khalf;
            v2f bf;
            bf.x = W1[(size_t)kb * 512 + ncol];
            bf.y = W1[(size_t)(kb + 1) * 512 + ncol];

            // A fragments from LDS: lane l15 -> row, khalf selects K pair
            const int krow = kk + 2 * khalf;
            v2f a0 = *(const v2f*)&ldsA[(l15 +  0) * LDS_STRIDE + krow];
            v2f a1 = *(const v2f*)&ldsA[(l15 + 16) * LDS_STRIDE + krow];
            v2f a2 = *(const v2f*)&ldsA[(l15 + 32) * LDS_STRIDE + krow];
            v2f a3 = *(const v2f*)&ldsA[(l15 + 48) * LDS_STRIDE + krow];

            c0 = __builtin_amdgcn_wmma_f32_16x16x4_f32(false, a0, false, bf, (short)0, c0, false, false);
            c1 = __builtin_amdgcn_wmma_f32_16x16x4_f32(false, a1, false, bf, (short)0, c1, false, false);
            c2 = __builtin_amdgcn_wmma_f32_16x16x4_f32(false, a2, false, bf, (short)0, c2, false, false);
            c3 = __builtin_amdgcn_wmma_f32_16x16x4_f32(false, a3, false, bf, (short)0, c3, false, false);
        }
        __syncthreads();
    }

    // Epilogue: bias + relu. C layout: VGPR r -> M = r + 8*khalf, N = l15.
    const float bias = b1[ncol];
    const int mbase  = M0 + 8 * khalf;
    #pragma unroll
    for (int rr = 0; rr < 8; ++rr) {
        float v;
        v = c0[rr] + bias; emb[(size_t)(mbase +  0 + rr) * 512 + ncol] = v > 0.f ? v : 0.f;
        v = c1[rr] + bias; emb[(size_t)(mbase + 16 + rr) * 512 + ncol] = v > 0.f ? v : 0.f;
        v = c2[rr] + bias; emb[(size_t)(mbase + 32 + rr) * 512 + ncol] = v > 0.f ? v : 0.f;
        v = c3[rr] + bias; emb[(size_t)(mbase + 48 + rr) * 512 + ncol] = v > 0.f ? v : 0.f;
    }
}

// ---------------------------------------------------------------------------
// Kernel 2: logits[b, j] = emb.reshape(B, 5120) @ W2 + b2   (N=10 padded to 16)
// One wave per 16-row M tile; masked (but unconditionally executed) B loads
// keep EXEC all-ones for WMMA.
// ---------------------------------------------------------------------------
__global__ __launch_bounds__(256) void k2_gemm2(
    const float* __restrict__ emb,   // [4096, 5120]
    const float* __restrict__ W2,    // [5120, 10]
    const float* __restrict__ b2,    // [10]
    float* __restrict__ logits)      // [4096, 10]
{
    const int tid     = threadIdx.x;
    const int wave    = tid >> 5;
    const int lane_id = tid & 31;
    const int khalf   = lane_id >> 4;
    const int n       = lane_id & 15;
    const int Mbase   = (blockIdx.x * 8 + wave) * 16;

    const int   nc    = n < 10 ? n : 9;
    const float nmask = n < 10 ? 1.f : 0.f;

    const float* arow = emb + (size_t)(Mbase + n) * K2;  // lane's A row
    v8f c = {};

    #pragma unroll 8
    for (int k = 0; k < K2; k += 4) {
        const int kA = k + 2 * khalf;
        v2f af = *(const v2f*)(arow + kA);               // contiguous b64
        v2f bf;
        bf.x = W2[(size_t)kA * 10 + nc] * nmask;
        bf.y = W2[(size_t)(kA + 1) * 10 + nc] * nmask;
        c = __builtin_amdgcn_wmma_f32_16x16x4_f32(false, af, false, bf, (short)0, c, false, false);
    }

    if (n < 10) {
        const float bias = b2[n];
        #pragma unroll
        for (int rr = 0; rr < 8; ++rr) {
            const int m = Mbase + rr + 8 * khalf;
            logits[(size_t)m * 10 + n] = c[rr] + bias;
        }
    }
}

// ---------------------------------------------------------------------------
// Kernel 3: per-batch one-hot gather of the three 512-wide context rows.
// ---------------------------------------------------------------------------
__global__ __launch_bounds__(128) void k3_gather(
    const float* __restrict__ lane,
    const float* __restrict__ nghl,
    const float* __restrict__ ngh,
    const int* __restrict__ label,     // [4096,10] one-hot
    float* __restrict__ out_lane,
    float* __restrict__ out_nghl,
    float* __restrict__ out_ngh)
{
    __shared__ int sidx;
    const int b = blockIdx.x;
    if (threadIdx.x == 0) {
        int idx = 0, best = label[b * 10];
        #pragma unroll
        for (int p = 1; p < 10; ++p) {
            const int v = label[b * 10 + p];
            if (v > best) { best = v; idx = p; }
        }
        sidx = idx;
    }
    __syncthreads();
    const int idx = sidx;
    const size_t so = ((size_t)b * 10 + idx) * 512 + threadIdx.x * 4;
    const size_t dx = (size_t)b * 512 + threadIdx.x * 4;
    *(float4*)(out_lane + dx) = *(const float4*)(lane + so);
    *(float4*)(out_nghl + dx) = *(const float4*)(nghl + so);
    *(float4*)(out_ngh  + dx) = *(const float4*)(ngh  + so);
}

// ---------------------------------------------------------------------------
extern "C" void kernel_launch(void* const* d_in, const int* in_sizes, int n_in,
                              void* d_out, int out_size, void* d_ws, size_t ws_size,
                              hipStream_t stream) {
    const float* agent = (const float*)d_in[0];
    const float* lane  = (const float*)d_in[1];
    const float* nghl  = (const float*)d_in[2];
    const float* ngh   = (const float*)d_in[3];
    const int*   label = (const int*)d_in[4];
    const float* W1    = (const float*)d_in[5];
    const float* b1    = (const float*)d_in[6];
    const float* W2    = (const float*)d_in[7];
    const float* b2    = (const float*)d_in[8];

    float* out      = (float*)d_out;
    float* logits   = out;                       // [4096,10]
    float* out_lane = out + (size_t)NB * NP;     // [4096,512]
    float* out_nghl = out_lane + (size_t)NB * NLD;
    float* out_ngh  = out_nghl + (size_t)NB * NLD;

    float* emb = (float*)d_ws;                   // [40960,512] fp32, 84 MB

    k1_gemm1_relu<<<dim3(MROWS / 64, 4), 256, 0, stream>>>(
        agent, lane, nghl, ngh, W1, b1, emb);
    k2_gemm2<<<NB / (8 * 16), 256, 0, stream>>>(emb, W2, b2, logits);
    k3_gather<<<NB, 128, 0, stream>>>(lane, nghl, ngh, label,
                                      out_lane, out_nghl, out_ngh);
}